// Block_84602265797044
// MI455X (gfx1250) — compile-verified
//
#include <hip/hip_runtime.h>

// MI455X / gfx1250, wave32. fp32 pipeline.
// - Matrix ops: native V_WMMA_F32_16X16X4_F32 (D = A(16x4) x B(4x16) + C).
// - Operand staging: GLOBAL_LOAD_ASYNC_TO_LDS_B128 (ASYNCcnt) into LDS,
//   shared by the 8 waves of each block; B/A operands then come from ds_load.

typedef float v2f __attribute__((ext_vector_type(2)));
typedef float v8f __attribute__((ext_vector_type(8)));

#define WMMA_F32(a, b, c) \
  __builtin_amdgcn_wmma_f32_16x16x4_f32(false, (a), false, (b), (short)0, (c), false, false)

__device__ __forceinline__ void async_wait0() {
#if __has_builtin(__builtin_amdgcn_s_wait_asynccnt)
  __builtin_amdgcn_s_wait_asynccnt(0);
#else
  asm volatile("s_wait_asynccnt 0x0" ::: "memory");
#endif
}

// Copy `chunks` x 16B from global to LDS with async-to-LDS DMA; each of the
// 256 threads issues its share, waits on its own ASYNCcnt. Caller barriers.
__device__ __forceinline__ void async_stage(void* lds, const void* g, int chunks) {
  unsigned lbase = (unsigned)(uintptr_t)lds;                 // low 32 bits = LDS offset
  unsigned long long gbase = (unsigned long long)(uintptr_t)g;
  for (int c = threadIdx.x; c < chunks; c += 256) {
    unsigned loff = lbase + (unsigned)c * 16u;
    unsigned long long ga = gbase + (unsigned long long)c * 16ull;
    asm volatile("global_load_async_to_lds_b128 %0, %1, off"
                 :: "v"(loff), "v"(ga) : "memory");
  }
  async_wait0();
}

// ---------------------------------------------------------------------------
// Channel-mix GEMM: Y[b,c,n,d] = sum_m W[c,m] * X[((b*32+n)*64+m)*128+d]
// One block per (b,n): stage the 64x128 slab (32KB) in LDS, 8 waves cover
// all (c,d) 16x16 tiles, B operand from LDS.
// ---------------------------------------------------------------------------
__global__ __launch_bounds__(256) void gemm_nd_kernel(
    const float* __restrict__ W, const float* __restrict__ X,
    float* __restrict__ Y, int Crows) {
  __shared__ float sx[64 * 128];                 // 32 KB slab x[b,n,:,:]
  int bn = blockIdx.x;                           // 0..255
  async_stage(sx, X + (size_t)bn * 8192, 8192 * 4 / 16);
  __syncthreads();

  int wv = threadIdx.x >> 5;
  int lane = threadIdx.x & 31;
  int hi = lane >> 4, lm = lane & 15;
  int tilesC = Crows >> 4;
  int ntiles = tilesC * 8;
  int b_ = bn >> 5, n_ = bn & 31;

  for (int t = wv; t < ntiles; t += 8) {
    int td = t & 7, tc = t >> 3;
    const float* arow = W + (size_t)(tc * 16 + lm) * 64 + 2 * hi;
    const float* bcol = sx + td * 16 + lm;
    v8f acc = {};
    for (int k0 = 0; k0 < 64; k0 += 4) {
      v2f a = { arow[k0], arow[k0 + 1] };
      v2f b = { bcol[(k0 + 2 * hi) * 128], bcol[(k0 + 2 * hi + 1) * 128] };
      acc = WMMA_F32(a, b, acc);
    }
    int d = td * 16 + lm;
    for (int v = 0; v < 8; v++) {
      int c = tc * 16 + v + 8 * hi;
      Y[(((size_t)b_ * Crows + c) * 32 + n_) * 128 + d] = acc[v];
    }
  }
}

// ---------------------------------------------------------------------------
// Row-major GEMM: C[M,N] = act( A[MxK] @ Bm[NxK]^T ), act 0=none 1=relu 2=sigmoid.
// Block = one 16-row A slab (staged in LDS, <=64KB) x 8 N-tiles (one per wave).
// Grid: (M/16) * (N/16/8) blocks.
// ---------------------------------------------------------------------------
__global__ __launch_bounds__(256) void fc_gemm_kernel(
    const float* __restrict__ A, const float* __restrict__ Bm,
    float* __restrict__ C, int M, int N, int K, int act) {
  __shared__ float sA[16 * 1024];                // up to 64 KB (K<=1024)
  const int tilesN8 = (N >> 4) >> 3;             // N-tile groups of 8
  int tm = blockIdx.x / tilesN8;
  int tb = blockIdx.x % tilesN8;
  async_stage(sA, A + (size_t)tm * 16 * K, 16 * K * 4 / 16);
  __syncthreads();

  int wv = threadIdx.x >> 5;
  int lane = threadIdx.x & 31;
  int hi = lane >> 4, lm = lane & 15;
  int tn = tb * 8 + wv;

  const float* ar = sA + (size_t)lm * K + 2 * hi;                  // LDS, K-contig
  const float* br = Bm + (size_t)(tn * 16 + lm) * K + 2 * hi;      // global, K-contig

  v8f acc = {};
  for (int k0 = 0; k0 < K; k0 += 4) {
    v2f a = { ar[k0], ar[k0 + 1] };
    v2f b = { br[k0], br[k0 + 1] };
    acc = WMMA_F32(a, b, acc);
  }
  for (int v = 0; v < 8; v++) {
    int row = tm * 16 + v + 8 * hi;
    float val = acc[v];
    if (act == 1) val = fmaxf(val, 0.f);
    else if (act == 2) val = 1.f / (1.f + expf(-val));
    C[(size_t)row * N + tn * 16 + lm] = val;
  }
}

// ---------------------------------------------------------------------------
// Final GEMM + shortcut, transposed store:
// out[b,n,o,d] = sum_m W[o,m]*Z[((b*32+n)*64+m)*128+d] + SC[b,o,n,d]
// ---------------------------------------------------------------------------
__global__ __launch_bounds__(256) void final_gemm_kernel(
    const float* __restrict__ W, const float* __restrict__ X,
    const float* __restrict__ SC, float* __restrict__ OUT) {
  __shared__ float sx[64 * 128];
  int bn = blockIdx.x;
  async_stage(sx, X + (size_t)bn * 8192, 8192 * 4 / 16);
  __syncthreads();

  int wv = threadIdx.x >> 5;
  int lane = threadIdx.x & 31;
  int hi = lane >> 4, lm = lane & 15;
  int b_ = bn >> 5, n_ = bn & 31;

  for (int t = wv; t < 64; t += 8) {             // 8 c-tiles x 8 d-tiles
    int td = t & 7, tc = t >> 3;
    const float* arow = W + (size_t)(tc * 16 + lm) * 64 + 2 * hi;
    const float* bcol = sx + td * 16 + lm;
    v8f acc = {};
    for (int k0 = 0; k0 < 64; k0 += 4) {
      v2f a = { arow[k0], arow[k0 + 1] };
      v2f b = { bcol[(k0 + 2 * hi) * 128], bcol[(k0 + 2 * hi + 1) * 128] };
      acc = WMMA_F32(a, b, acc);
    }
    int d = td * 16 + lm;
    for (int v = 0; v < 8; v++) {
      int c = tc * 16 + v + 8 * hi;
      float val = acc[v] + SC[(((size_t)b_ * 128 + c) * 32 + n_) * 128 + d];
      OUT[(((size_t)b_ * 32 + n_) * 128 + c) * 128 + d] = val;
    }
  }
}

// ---------------------------------------------------------------------------
// Batchnorm stats: one block per channel c; layout Y[(b*C+c)*S + s].
// ---------------------------------------------------------------------------
__global__ __launch_bounds__(256) void bn_stats_kernel(
    const float* __restrict__ Y, float* __restrict__ mean, float* __restrict__ inv,
    int C, int S, int Bn) {
  int c = blockIdx.x, tid = threadIdx.x;
  float s1 = 0.f, s2 = 0.f;
  for (int b = 0; b < Bn; b++) {
    const float* p = Y + ((size_t)b * C + c) * S;
    for (int i = tid; i < S; i += 256) { float v = p[i]; s1 += v; s2 += v * v; }
  }
  __shared__ float r1[256], r2[256];
  r1[tid] = s1; r2[tid] = s2; __syncthreads();
  for (int off = 128; off > 0; off >>= 1) {
    if (tid < off) { r1[tid] += r1[tid + off]; r2[tid] += r2[tid + off]; }
    __syncthreads();
  }
  if (tid == 0) {
    float cnt = (float)Bn * (float)S;
    float m = r1[0] / cnt;
    float var = r2[0] / cnt - m * m;
    mean[c] = m;
    inv[c] = rsqrtf(var + 1e-5f);
  }
}

__global__ void bn_apply_kernel(float* __restrict__ Y,
    const float* __restrict__ mean, const float* __restrict__ inv,
    const float* __restrict__ g, const float* __restrict__ be,
    int C, int S, int total, int relu) {
  for (int idx = blockIdx.x * blockDim.x + threadIdx.x; idx < total;
       idx += gridDim.x * blockDim.x) {
    int c = (idx / S) % C;
    float v = g[c] * (Y[idx] - mean[c]) * inv[c] + be[c];
    if (relu) v = fmaxf(v, 0.f);
    Y[idx] = v;
  }
}

// ---------------------------------------------------------------------------
// Attention prep, one block per (b,m): from a(32x128) compute
//   adj[i,j] = xc.xc / (|xc_i||xc_j| + 1e-8)
//   ef[i,j]  = 0.5*(mean_i + 2*mean_j - (sum_d a_j[d]*[a_i[d]==a_j[d]])/128)
// ---------------------------------------------------------------------------
__global__ __launch_bounds__(256) void attn_prep_kernel(
    const float* __restrict__ A, float* __restrict__ ef, float* __restrict__ adj) {
  int bm = blockIdx.x, tid = threadIdx.x;
  __shared__ float sa[32 * 128];
  __shared__ float part[256];
  __shared__ float smean[32], snrm[32];
  async_stage(sa, A + (size_t)bm * 4096, 4096 * 4 / 16);
  __syncthreads();

  int r = tid >> 3, q = tid & 7;                 // 8 threads per row
  float ps = 0.f;
  for (int d = q * 16; d < q * 16 + 16; d++) ps += sa[r * 128 + d];
  part[tid] = ps; __syncthreads();
  if (q == 0) { float s = 0.f; for (int k = 0; k < 8; k++) s += part[tid + k];
                smean[r] = s * (1.f / 128.f); }
  __syncthreads();
  float m = smean[r]; ps = 0.f;
  for (int d = q * 16; d < q * 16 + 16; d++) { float v = sa[r * 128 + d] - m; ps += v * v; }
  part[tid] = ps; __syncthreads();
  if (q == 0) { float s = 0.f; for (int k = 0; k < 8; k++) s += part[tid + k];
                snrm[r] = sqrtf(s); }
  __syncthreads();

  for (int k = 0; k < 4; k++) {
    int p = tid * 4 + k;
    int i = p >> 5, j = p & 31;
    float mi = smean[i], mj = smean[j];
    float num = 0.f, eqs = 0.f;
    for (int d = 0; d < 128; d++) {
      float ai = sa[i * 128 + d], aj = sa[j * 128 + d];
      num += (ai - mi) * (aj - mj);
      eqs += (ai == aj) ? aj : 0.f;
    }
    adj[(size_t)bm * 1024 + p] = num / (snrm[i] * snrm[j] + 1e-8f);
    ef[(size_t)bm * 1024 + p]  = 0.5f * (mi + 2.f * mj - eqs * (1.f / 128.f));
  }
}

// ---------------------------------------------------------------------------
// Masked softmax + apply: att = softmax(adj>0 ? e2 : -1e12); out = att @ a
// ---------------------------------------------------------------------------
__global__ __launch_bounds__(256) void softmax_apply_kernel(
    const float* __restrict__ A, const float* __restrict__ adj,
    const float* __restrict__ e2, float* __restrict__ out) {
  int bm = blockIdx.x, tid = threadIdx.x;
  __shared__ float sa[4096];
  __shared__ float satt[1024];
  async_stage(sa, A + (size_t)bm * 4096, 4096 * 4 / 16);
  for (int p = tid; p < 1024; p += 256) {
    float av = adj[(size_t)bm * 1024 + p];
    satt[p] = (av > 0.f) ? e2[(size_t)bm * 1024 + p] : -1e12f;
  }
  __syncthreads();
  if (tid < 32) {
    int i = tid;
    float mx = -3.4e38f;
    for (int j = 0; j < 32; j++) mx = fmaxf(mx, satt[i * 32 + j]);
    float s = 0.f;
    for (int j = 0; j < 32; j++) { float e = expf(satt[i * 32 + j] - mx);
                                   satt[i * 32 + j] = e; s += e; }
    float is = 1.f / s;
    for (int j = 0; j < 32; j++) satt[i * 32 + j] *= is;
  }
  __syncthreads();
  for (int o = tid; o < 4096; o += 256) {
    int i = o >> 7, d = o & 127;
    float s = 0.f;
    for (int j = 0; j < 32; j++) s += satt[i * 32 + j] * sa[j * 128 + d];
    out[(size_t)bm * 4096 + o] = s;
  }
}

// ---------------------------------------------------------------------------
// 3x3 depthwise conv over (M=64, D=128) spatial, channel n (32), pad 1.
// Input Z layout (B,64,32,128) [b,m,n,d]; output (B,32,64,128) [b,n,m,d].
// ---------------------------------------------------------------------------
__global__ void dwconv_kernel(const float* __restrict__ Z,
    const float* __restrict__ Wd, float* __restrict__ out, int total) {
  for (int idx = blockIdx.x * blockDim.x + threadIdx.x; idx < total;
       idx += gridDim.x * blockDim.x) {
    int d = idx & 127;
    int m = (idx >> 7) & 63;
    int n = (idx >> 13) & 31;
    int b = idx >> 18;
    float s = 0.f;
    for (int kh = 0; kh < 3; kh++) {
      int mm = m + kh - 1; if (mm < 0 || mm >= 64) continue;
      for (int kw = 0; kw < 3; kw++) {
        int dd = d + kw - 1; if (dd < 0 || dd >= 128) continue;
        s += Wd[n * 9 + kh * 3 + kw] * Z[(((size_t)b * 64 + mm) * 32 + n) * 128 + dd];
      }
    }
    out[idx] = s;
  }
}

// ---------------------------------------------------------------------------
extern "C" void kernel_launch(void* const* d_in, const int* in_sizes, int n_in,
                              void* d_out, int out_size, void* d_ws, size_t ws_size,
                              hipStream_t stream) {
  const float* x     = (const float*)d_in[0];
  const float* W_sc  = (const float*)d_in[1];
  const float* g_sc  = (const float*)d_in[2];
  const float* b_sc  = (const float*)d_in[3];
  const float* W_l1  = (const float*)d_in[4];
  const float* g1    = (const float*)d_in[5];
  const float* b1    = (const float*)d_in[6];
  const float* W_fc1 = (const float*)d_in[7];
  const float* W_fc2 = (const float*)d_in[8];
  const float* g2    = (const float*)d_in[9];
  const float* b2    = (const float*)d_in[10];
  const float* W_dw  = (const float*)d_in[11];
  const float* g3    = (const float*)d_in[12];
  const float* b3    = (const float*)d_in[13];
  const float* W_l3  = (const float*)d_in[14];
  float* out = (float*)d_out;

  // Workspace layout (floats), total ~12.32M floats (~49.3 MB)
  float* ws  = (float*)d_ws;
  float* ysc = ws;                   // 4,194,304 : (B,128,32,128) shortcut path
  float* y1  = ysc + 4194304;        // 2,097,152 : (B,64,32,128)  main path
  float* ef  = y1  + 2097152;        //   524,288 : (512,1024)
  float* adj = ef  + 524288;         //   524,288
  float* h   = adj + 524288;         //   262,144 : (512,512)
  float* e2  = h   + 262144;         //   524,288 : (512,1024)
  float* att = e2  + 524288;         // 2,097,152 : (B,64,32,128)
  float* cv  = att + 2097152;        // 2,097,152 : (B,32,64,128)
  float* st  = cv  + 2097152;        //     1,024 : BN stat buffers
  float *m1 = st,       *i1 = st + 128;
  float *m2 = st + 256, *i2 = st + 384;
  float *m3 = st + 512, *i3 = st + 640;
  float *m4 = st + 768, *i4 = st + 896;

  // 1) channel-mix GEMMs (WMMA f32, LDS-staged via async-to-LDS)
  gemm_nd_kernel<<<256, 256, 0, stream>>>(W_l1, x, y1, 64);
  gemm_nd_kernel<<<256, 256, 0, stream>>>(W_sc, x, ysc, 128);

  // 2) batchnorms (in place)
  bn_stats_kernel<<<64, 256, 0, stream>>>(y1, m1, i1, 64, 4096, 8);
  bn_apply_kernel<<<2048, 256, 0, stream>>>(y1, m1, i1, g1, b1, 64, 4096, 2097152, 0);
  bn_stats_kernel<<<128, 256, 0, stream>>>(ysc, m2, i2, 128, 4096, 8);
  bn_apply_kernel<<<2048, 256, 0, stream>>>(ysc, m2, i2, g_sc, b_sc, 128, 4096, 4194304, 0);

  // 3) attention: corr/adj + e features, fc1/fc2 GEMMs (WMMA), softmax apply
  attn_prep_kernel<<<512, 256, 0, stream>>>(y1, ef, adj);
  fc_gemm_kernel<<<128, 256, 0, stream>>>(ef, W_fc1, h, 512, 512, 1024, /*relu*/1);
  fc_gemm_kernel<<<256, 256, 0, stream>>>(h, W_fc2, e2, 512, 1024, 512, /*sigmoid*/2);
  softmax_apply_kernel<<<512, 256, 0, stream>>>(y1, adj, e2, att);

  // 4) bn2 + relu (in place)
  bn_stats_kernel<<<64, 256, 0, stream>>>(att, m3, i3, 64, 4096, 8);
  bn_apply_kernel<<<2048, 256, 0, stream>>>(att, m3, i3, g2, b2, 64, 4096, 2097152, 1);

  // 5) depthwise 3x3 conv + bn3 + relu (in place)
  dwconv_kernel<<<2048, 256, 0, stream>>>(att, W_dw, cv, 2097152);
  bn_stats_kernel<<<32, 256, 0, stream>>>(cv, m4, i4, 32, 8192, 8);
  bn_apply_kernel<<<2048, 256, 0, stream>>>(cv, m4, i4, g3, b3, 32, 8192, 2097152, 1);

  // 6) final GEMM (WMMA) + shortcut, transposed store to d_out
  final_gemm_kernel<<<256, 256, 0, stream>>>(W_l3, cv, ysc, out);
}